// ExpmLogm_54872502174212
// MI455X (gfx1250) — compile-verified
//
#include <hip/hip_runtime.h>
#include <math.h>

#define PLANE   (128*128*128)   // voxels per batch (D*H*W)
#define TPB     256             // 8 waves (wave32)
#define VPT     4               // voxels per thread (128-bit streams)
#define TILE    (TPB*VPT)       // 1024 voxels per block-tile

typedef float v4f __attribute__((ext_vector_type(4)));   // native 128-bit vector

// f(x) = log(exp(x)); fast hw approximations suffice: result feeds a ~1e-7 correction.
__device__ __forceinline__ float flog_exp(float x) { return __logf(__expf(x)); }

// out = A + sum_i (f(l_i)-l_i) * u_i u_i^T  (numerically robust spectral map, branchless)
__device__ __forceinline__ void sym3_process(
    float a00, float a01, float a02, float a11, float a12, float a22,
    float& o00, float& o01, float& o02, float& o11, float& o12, float& o22)
{
    const float q   = (a00 + a11 + a22) * (1.0f/3.0f);
    const float b00 = a00 - q, b11 = a11 - q, b22 = a22 - q;
    const float p1  = a01*a01 + a02*a02 + a12*a12;
    // Clamp instead of branching: degenerate (A ~ q*I) lanes get p ~ 4e-15, so the
    // trig path yields l ~ q to within 1e-14 and stays uniform across the wave.
    const float p2  = fmaxf(b00*b00 + b11*b11 + b22*b22 + 2.0f*p1, 1e-28f);

    o00 = a00; o01 = a01; o02 = a02; o11 = a11; o12 = a12; o22 = a22;

    const float p  = __fsqrt_rn(p2 * (1.0f/6.0f));
    const float ip = __frcp_rn(p);
    const float c00 = b00*ip, c11 = b11*ip, c22 = b22*ip;
    const float c01 = a01*ip, c02 = a02*ip, c12 = a12*ip;
    const float det = c00*(c11*c22 - c12*c12)
                    - c01*(c01*c22 - c12*c02)
                    + c02*(c01*c12 - c11*c02);
    const float r   = fminf(1.0f, fmaxf(-1.0f, 0.5f*det));
    const float phi = acosf(r) * (1.0f/3.0f);
    const float tp  = 2.0f*p;
    const float l0 = q + tp*__cosf(phi);
    const float l2 = q + tp*__cosf(phi + 2.0943951023931953f);  // + 2*pi/3
    const float l1 = 3.0f*q - l0 - l2;
    const float lam3[3] = { l0, l1, l2 };

    #pragma unroll
    for (int k = 0; k < 3; ++k) {
        const float lam = lam3[k];
        const float d   = flog_exp(lam) - lam;          // ~1e-7 correction
        // eigenvector of (A - lam I): largest cross product of row pairs
        const float r0x = a00 - lam, r0y = a01,       r0z = a02;
        const float r1x = a01,       r1y = a11 - lam, r1z = a12;
        const float r2x = a02,       r2y = a12,       r2z = a22 - lam;
        const float v0x = r0y*r1z - r0z*r1y, v0y = r0z*r1x - r0x*r1z, v0z = r0x*r1y - r0y*r1x;
        const float v1x = r0y*r2z - r0z*r2y, v1y = r0z*r2x - r0x*r2z, v1z = r0x*r2y - r0y*r2x;
        const float v2x = r1y*r2z - r1z*r2y, v2y = r1z*r2x - r1x*r2z, v2z = r1x*r2y - r1y*r2x;
        const float n0 = v0x*v0x + v0y*v0y + v0z*v0z;
        const float n1 = v1x*v1x + v1y*v1y + v1z*v1z;
        const float n2 = v2x*v2x + v2y*v2y + v2z*v2z;
        float ux = v0x, uy = v0y, uz = v0z, nn = n0;   // per-lane selects -> v_cndmask
        if (n1 > nn) { ux = v1x; uy = v1y; uz = v1z; nn = n1; }
        if (n2 > nn) { ux = v2x; uy = v2y; uz = v2z; nn = n2; }
        const float w = (nn > 1e-30f) ? (d * __frcp_rn(nn)) : 0.0f; // d * (v v^T)/||v||^2
        o00 += w*ux*ux; o01 += w*ux*uy; o02 += w*ux*uz;
        o11 += w*uy*uy; o12 += w*uy*uz; o22 += w*uz*uz;
    }
}

__device__ __forceinline__ unsigned lds_byte_off(const void* p) {
    // generic pointer into LDS aperture: low 32 bits are the LDS byte address
    return (unsigned)(unsigned long long)(uintptr_t)p;
}

__global__ __launch_bounds__(TPB)
void ExpmLogm_54872502174212_kernel(const float* __restrict__ x,
                                    float* __restrict__ y,
                                    int ntiles, int tilesPerBatch)
{
    __shared__ float smem[6 * TILE];   // 24 KB: 6 symmetric-unique channel tiles
    const int tid = threadIdx.x;
    constexpr int cin[6] = {0, 1, 2, 4, 5, 8};   // unique channels of symmetric 3x3

    for (int t = blockIdx.x; t < ntiles; t += gridDim.x) {
        const int b = t / tilesPerBatch;
        const long long v0   = (long long)(t - b * tilesPerBatch) * TILE;
        const long long base = ((long long)b * 9) * PLANE + v0 + (long long)tid * VPT;

        // Ensure previous iteration's LDS reads have drained before DMA overwrites slots.
        asm volatile("s_wait_dscnt 0" ::: "memory");

        // Async DMA: global -> LDS, 128b per lane per channel (ASYNCcnt-tracked).
        // NT hint: pure stream, zero reuse -> keep it out of L2's useful set.
        #pragma unroll
        for (int s = 0; s < 6; ++s) {
            const float* g = x + base + (long long)cin[s] * PLANE;
            const unsigned long long ga = (unsigned long long)(uintptr_t)g;
            const unsigned lo = lds_byte_off(&smem[s * TILE + tid * VPT]);
            asm volatile("global_load_async_to_lds_b128 %0, %1, off th:TH_LOAD_NT"
                         :: "v"(lo), "v"(ga) : "memory");
        }
        asm volatile("s_wait_asynccnt 0" ::: "memory");

        // Pull staged tile into registers (ds_load_b128, same-lane slots: no barrier).
        float va[6][4];
        #pragma unroll
        for (int s = 0; s < 6; ++s)
            *(v4f*)va[s] = *(const v4f*)&smem[s * TILE + tid * VPT];

        float vo[6][4];
        #pragma unroll
        for (int i = 0; i < VPT; ++i) {
            sym3_process(va[0][i], va[1][i], va[2][i], va[3][i], va[4][i], va[5][i],
                         vo[0][i], vo[1][i], vo[2][i], vo[3][i], vo[4][i], vo[5][i]);
        }

        // Write all 9 channel planes (duplicating symmetric off-diagonals), NT 128b stores.
        float* yb = y + base;
        const int omap[9] = {0, 1, 2, 1, 3, 4, 2, 4, 5};  // o00 o01 o02 o01 o11 o12 o02 o12 o22
        #pragma unroll
        for (int c = 0; c < 9; ++c) {
            v4f v = *(const v4f*)vo[omap[c]];
            __builtin_nontemporal_store(v, (v4f*)(yb + (long long)c * PLANE));
        }
    }
}

extern "C" void kernel_launch(void* const* d_in, const int* in_sizes, int n_in,
                              void* d_out, int out_size, void* d_ws, size_t ws_size,
                              hipStream_t stream) {
    const float* x = (const float*)d_in[0];
    float*       y = (float*)d_out;

    const long long nvox          = (long long)in_sizes[0] / 9;    // B * D*H*W
    const int       ntiles        = (int)(nvox / TILE);            // 4096 for B=2
    const int       tilesPerBatch = PLANE / TILE;                  // 2048

    const int grid = ntiles;  // one tile per block; grid-stride loop handles any mismatch
    ExpmLogm_54872502174212_kernel<<<grid, TPB, 0, stream>>>(x, y, ntiles, tilesPerBatch);
}